// FDA_37434934952153
// MI455X (gfx1250) — compile-verified
//
#include <hip/hip_runtime.h>
#include <math.h>

typedef float v2f __attribute__((ext_vector_type(2)));
typedef float v8f __attribute__((ext_vector_type(8)));

#define BATCH 16
#define CIN   128
#define HW    24
#define SP    576      // 24*24
#define NGRP  4

// ---------------------------------------------------------------------------
// fp32 WMMA: D(16x16) += A(16x4) * B(4x16), full fp32 accumulation.
// A layout: lanes 0-15 -> M=lane, {K0,K1}; lanes 16-31 -> M=lane-16, {K2,K3}
// B layout: lanes 0-15 -> N=lane, {K0,K1}; lanes 16-31 -> N=lane-16, {K2,K3}
// D layout: vgpr r: lanes 0-15 -> (M=r, N=lane); lanes 16-31 -> (M=r+8, N=lane-16)
// ---------------------------------------------------------------------------
__device__ __forceinline__ v8f wmma4(float a0, float a1, float b0, float b1, v8f acc) {
  v2f a = {a0, a1};
  v2f b = {b0, b1};
  return __builtin_amdgcn_wmma_f32_16x16x4_f32(false, a, false, b, (short)0, acc,
                                               false, false);
}

// ---------------------------------------------------------------------------
// avg_ref[b][c] = mean over H*W of ref
// ---------------------------------------------------------------------------
__global__ void k_avg_ref(const float* __restrict__ ref, float* __restrict__ avg) {
  int i = blockIdx.x * blockDim.x + threadIdx.x;
  if (i >= BATCH * CIN) return;
  const float* p = ref + (size_t)i * SP;
  float s = 0.f;
  for (int j = 0; j < SP; ++j) s += p[j];
  avg[i] = s * (1.0f / SP);
}

// ---------------------------------------------------------------------------
// L2-normalize x and ref along channels at every spatial position.
// Output transposed to [b][pos][c] so GEMM K-pairs are contiguous (b64 loads).
// ---------------------------------------------------------------------------
__global__ void k_normalize(const float* __restrict__ x, const float* __restrict__ ref,
                            float* __restrict__ qn, float* __restrict__ sn) {
  int bp = blockIdx.x;            // b*SP + pos
  int b = bp / SP, pos = bp % SP;
  int c = threadIdx.x;            // 128 threads
  float vx = x[((size_t)b * CIN + c) * SP + pos];
  float vr = ref[((size_t)b * CIN + c) * SP + pos];
  __shared__ float sx[128], sr[128];
  sx[c] = vx * vx; sr[c] = vr * vr;
  __syncthreads();
  for (int off = 64; off > 0; off >>= 1) {
    if (c < off) { sx[c] += sx[c + off]; sr[c] += sr[c + off]; }
    __syncthreads();
  }
  float inx = 1.0f / (sqrtf(sx[0]) + 1e-5f);
  float inr = 1.0f / (sqrtf(sr[0]) + 1e-5f);
  qn[((size_t)b * SP + pos) * CIN + c] = vx * inx;
  sn[((size_t)b * SP + pos) * CIN + c] = vr * inr;
}

// ---------------------------------------------------------------------------
// corr[b][i][j] = relu( sum_c qn[b][i][c] * sn[b][j][c] ), 576x576x128 per batch
// ---------------------------------------------------------------------------
__global__ void k_corr_wmma(const float* __restrict__ qn, const float* __restrict__ sn,
                            float* __restrict__ corr) {
  const int NT = SP / 16;                                // 36
  int wave = blockIdx.x * (blockDim.x >> 5) + (threadIdx.x >> 5);
  int lane = threadIdx.x & 31;
  if (wave >= BATCH * NT * NT) return;
  int jt = wave % NT;
  int it = (wave / NT) % NT;
  int b  = wave / (NT * NT);
  int hf = lane >> 4, l = lane & 15;
  const float* qp = qn + ((size_t)b * SP + it * 16 + l) * CIN + hf * 2;
  const float* sp = sn + ((size_t)b * SP + jt * 16 + l) * CIN + hf * 2;
  v8f acc = {};
  #pragma unroll 8
  for (int k = 0; k < CIN; k += 4)
    acc = wmma4(qp[k], qp[k + 1], sp[k], sp[k + 1], acc);
  float* op = corr + ((size_t)b * SP + it * 16) * SP + jt * 16;
  #pragma unroll
  for (int r = 0; r < 8; ++r) {
    float v = acc[r];
    op[(size_t)(r + hf * 8) * SP + l] = v > 0.f ? v : 0.f;
  }
}

// ---------------------------------------------------------------------------
// Block-1 center-pivot conv4d (ic=1 -> oc=16, stride 2). VALU (A-frag would be
// 15/16 empty in WMMA). Output channel-LAST: t1[b][n][oc], n=(ha,wa,hb',wb'),
// so each thread writes 16 contiguous floats (-> b128 stores).
// ---------------------------------------------------------------------------
__global__ void k_conv4d_b1(const float* __restrict__ corr,
                            const float* __restrict__ w1, const float* __restrict__ b1,
                            const float* __restrict__ w2, const float* __restrict__ b2,
                            float* __restrict__ out) {
  __shared__ float lw1[144], lw2[144], lb[16];
  int t = threadIdx.x;
  if (t < 144) { lw1[t] = w1[t]; lw2[t] = w2[t]; }
  if (t < 16) lb[t] = b1[t] + b2[t];
  __syncthreads();
  int idx = blockIdx.x * blockDim.x + t;               // B*576*144
  if (idx >= BATCH * SP * 144) return;
  int wbp = idx % 12;
  int hbp = (idx / 12) % 12;
  int wa  = (idx / 144) % HW;
  int ha  = (idx / (144 * HW)) % HW;
  int b   = idx / (144 * SP);
  const float* cb = corr + (size_t)b * SP * SP;
  float v1[9], v2[9];
  int jb = (2 * hbp) * HW + 2 * wbp;                   // branch1: pruned (hb,wb)
  int ia = ha * HW + wa;                               // branch2: fixed (ha,wa)
  #pragma unroll
  for (int dy = 0; dy < 3; ++dy)
    #pragma unroll
    for (int dx = 0; dx < 3; ++dx) {
      int hy = ha + dy - 1, wx = wa + dx - 1;
      v1[dy * 3 + dx] = (hy >= 0 && hy < HW && wx >= 0 && wx < HW)
                        ? cb[(size_t)(hy * HW + wx) * SP + jb] : 0.f;
      int hb = 2 * hbp + dy - 1, wb = 2 * wbp + dx - 1;
      v2[dy * 3 + dx] = (hb >= 0 && hb < HW && wb >= 0 && wb < HW)
                        ? cb[(size_t)ia * SP + hb * HW + wb] : 0.f;
    }
  float* op = out + (size_t)idx * 16;                  // channel-last
  #pragma unroll
  for (int oc = 0; oc < 16; ++oc) {
    float acc = lb[oc];
    #pragma unroll
    for (int q = 0; q < 9; ++q) acc += lw1[oc * 9 + q] * v1[q] + lw2[oc * 9 + q] * v2[q];
    op[oc] = acc;
  }
}

// ---------------------------------------------------------------------------
// Repack conv4d weights (oc,ic,3,3)x2 -> wT[br][tap][oc][ic] so the A-fragment
// K-pair (ic, ic+1) is one contiguous b64 load.
// ---------------------------------------------------------------------------
template <int IC, int OC>
__global__ void k_repack_w(const float* __restrict__ w1, const float* __restrict__ w2,
                           float* __restrict__ wT) {
  int i = blockIdx.x * blockDim.x + threadIdx.x;       // 2*9*OC*IC
  if (i >= 2 * 9 * OC * IC) return;
  int ic  = i % IC;
  int oc  = (i / IC) % OC;
  int tap = (i / (IC * OC)) % 9;
  int br  = i / (IC * OC * 9);
  const float* w = br ? w2 : w1;
  wT[i] = w[((size_t)oc * IC + ic) * 9 + tap];
}

// ---------------------------------------------------------------------------
// GroupNorm (channel-LAST tensors [b][n][c]): stats per (b, group), then fused
// normalize+ReLU in place.
// ---------------------------------------------------------------------------
__global__ void k_gn_stats_cl(const float* __restrict__ x, int C, int nsp,
                              float* __restrict__ stats) {
  int bg = blockIdx.x;                                 // B*NGRP
  int b = bg / NGRP, g = bg % NGRP;
  int cpg = C / NGRP;
  const float* p = x + (size_t)b * nsp * C + g * cpg;
  size_t n = (size_t)nsp * cpg;
  double s = 0.0, ss = 0.0;
  for (size_t i = threadIdx.x; i < n; i += blockDim.x) {
    size_t row = i / cpg; int j = (int)(i % cpg);
    float v = p[row * C + j];
    s += v; ss += (double)v * v;
  }
  __shared__ double ls[256], lss[256];
  int t = threadIdx.x;
  ls[t] = s; lss[t] = ss;
  __syncthreads();
  for (int off = 128; off > 0; off >>= 1) {
    if (t < off) { ls[t] += ls[t + off]; lss[t] += lss[t + off]; }
    __syncthreads();
  }
  if (t == 0) {
    double m = ls[0] / (double)n;
    double var = lss[0] / (double)n - m * m;
    stats[bg * 2]     = (float)m;
    stats[bg * 2 + 1] = (float)(1.0 / sqrt(var + 1e-5));
  }
}

__global__ void k_gn_apply_cl(float* __restrict__ x, int C,
                              const float* __restrict__ gamma, const float* __restrict__ beta,
                              const float* __restrict__ stats, size_t per_b, size_t total) {
  size_t i = (size_t)blockIdx.x * blockDim.x + threadIdx.x;
  if (i >= total) return;
  int c = (int)(i % C);                                // channel-last: trivial
  int b = (int)(i / per_b);
  int g = c / (C / NGRP);
  float m = stats[(b * NGRP + g) * 2], r = stats[(b * NGRP + g) * 2 + 1];
  float v = (x[i] - m) * r * gamma[c] + beta[c];
  x[i] = v > 0.f ? v : 0.f;                            // fused ReLU
}

// ---------------------------------------------------------------------------
// Center-pivot conv4d blocks 2/3 as WMMA GEMM, channel-last activations.
// D[oc, n] with n flattened over (ha,wa,hb',wb'); K = 2 branches x 9 taps x IC.
// in  [b][n_in][ic]  (n_in over (ha,wa,hb,wb)),  out [b][n][oc].
// Per WMMA: one b64 A load + one b64 B load per lane; D stores are contiguous.
// ---------------------------------------------------------------------------
template <int IC, int OC, int HB_IN, int STRIDE>
__global__ void k_conv4d_wmma(const float* __restrict__ in, float* __restrict__ out,
                              const float* __restrict__ wT,
                              const float* __restrict__ b1, const float* __restrict__ b2) {
  constexpr int HB_OUT = HB_IN / STRIDE;
  constexpr int BBOUT = HB_OUT * HB_OUT;
  constexpr int BBIN  = HB_IN * HB_IN;
  constexpr int NSP = SP * BBOUT;
  constexpr int NT = NSP / 16;
  constexpr int MT = OC / 16;
  int wave = blockIdx.x * (blockDim.x >> 5) + (threadIdx.x >> 5);
  int lane = threadIdx.x & 31;
  if (wave >= BATCH * MT * NT) return;
  int nt = wave % NT;
  int mt = (wave / NT) % MT;
  int b  = wave / (NT * MT);
  int hf = lane >> 4, l = lane & 15;
  int n = nt * 16 + l;                                 // this lane's N (B frag)
  int wbp = n % HB_OUT;
  int hbp = (n / HB_OUT) % HB_OUT;
  int wa  = (n / BBOUT) % HW;
  int ha  = n / (BBOUT * HW);
  int m = mt * 16 + l;                                 // this lane's M (A frag)
  int koff = hf * 2;
  const float* inb = in + (size_t)b * SP * BBIN * IC + koff;
  v8f acc = {};
  for (int br = 0; br < 2; ++br) {
    for (int tap = 0; tap < 9; ++tap) {
      int dy = tap / 3, dx = tap % 3;
      int ia, ja, ib, jb;
      if (br == 0) { ia = ha + dy - 1; ja = wa + dx - 1; ib = STRIDE * hbp;          jb = STRIDE * wbp; }
      else         { ia = ha;          ja = wa;          ib = STRIDE * hbp + dy - 1; jb = STRIDE * wbp + dx - 1; }
      bool valid = (ia >= 0) & (ia < HW) & (ja >= 0) & (ja < HW) &
                   (ib >= 0) & (ib < HB_IN) & (jb >= 0) & (jb < HB_IN);
      long long base = (long long)((ia * HW + ja) * BBIN + ib * HB_IN + jb) * IC;
      const float* ap = wT + ((size_t)(br * 9 + tap) * OC + m) * IC + koff;
      #pragma unroll
      for (int ic0 = 0; ic0 < IC; ic0 += 4) {
        float a0 = ap[ic0];                            // contiguous -> b64
        float a1 = ap[ic0 + 1];
        float v0 = valid ? inb[base + ic0] : 0.f;      // contiguous -> b64
        float v1 = valid ? inb[base + ic0 + 1] : 0.f;
        acc = wmma4(a0, a1, v0, v1, acc);
      }
    }
  }
  int oc0 = mt * 16 + hf * 8;                          // 8 contiguous oc per lane
  float* op = out + ((size_t)b * NSP + n) * OC + oc0;
  #pragma unroll
  for (int r = 0; r < 8; ++r) op[r] = acc[r] + b1[oc0 + r] + b2[oc0 + r];
}

// ---------------------------------------------------------------------------
// corrm[b][c][pos] = mean over 36 (hb',wb') of channel-last t3[b][pos*36+bb][c]
// (also transposes back to channel-first for the pointwise GEMMs)
// ---------------------------------------------------------------------------
__global__ void k_mean_bb_cl(const float* __restrict__ t3, float* __restrict__ corrm) {
  int bp = blockIdx.x;                                 // b*SP + pos
  int b = bp / SP, pos = bp % SP;
  int c = threadIdx.x;                                 // 128 threads, coalesced
  const float* p = t3 + ((size_t)b * SP + pos) * 36 * CIN + c;
  float s = 0.f;
  #pragma unroll
  for (int k = 0; k < 36; ++k) s += p[(size_t)k * CIN];
  corrm[((size_t)b * CIN + c) * SP + pos] = s * (1.0f / 36.0f);
}

// ---------------------------------------------------------------------------
// Pointwise (1x1) conv GEMMs: M=64, K=128, N=576 per batch, fused B-operand
// prep.  MODE 0: plain(corrm)  1: |bcast(p0)-p1|  2: |p0-p1|
//        3: concat(p0[64], p1[64])
// ---------------------------------------------------------------------------
template <int MODE>
__device__ __forceinline__ float pw_fetch(const float* __restrict__ p0,
                                          const float* __restrict__ p1,
                                          int b, int k, int n) {
  if (MODE == 0) return p0[((size_t)b * CIN + k) * SP + n];
  if (MODE == 1) return fabsf(p0[(size_t)b * CIN + k] - p1[((size_t)b * CIN + k) * SP + n]);
  if (MODE == 2) return fabsf(p0[((size_t)b * CIN + k) * SP + n] - p1[((size_t)b * CIN + k) * SP + n]);
  return (k < 64) ? p0[((size_t)b * 64 + k) * SP + n]
                  : p1[((size_t)b * 64 + (k - 64)) * SP + n];
}

template <int MODE>
__global__ void k_pw(const float* __restrict__ w, const float* __restrict__ bias,
                     const float* __restrict__ p0, const float* __restrict__ p1,
                     float* __restrict__ out) {
  const int NT = 36, MT = 4;
  int wave = blockIdx.x * (blockDim.x >> 5) + (threadIdx.x >> 5);
  int lane = threadIdx.x & 31;
  if (wave >= BATCH * MT * NT) return;
  int nt = wave % NT, mt = (wave / NT) % MT, b = wave / (NT * MT);
  int hf = lane >> 4, l = lane & 15;
  int n = nt * 16 + l;
  int m = mt * 16 + l;
  int koff = hf * 2;
  v8f acc = {};
  #pragma unroll 8
  for (int k0 = 0; k0 < 128; k0 += 4) {
    int k = k0 + koff;
    float a0 = w[(size_t)m * 128 + k];
    float a1 = w[(size_t)m * 128 + k + 1];
    float v0 = pw_fetch<MODE>(p0, p1, b, k, n);
    float v1 = pw_fetch<MODE>(p0, p1, b, k + 1, n);
    acc = wmma4(a0, a1, v0, v1, acc);
  }
  #pragma unroll
  for (int r = 0; r < 8; ++r) {
    int mm = mt * 16 + r + hf * 8;
    out[((size_t)b * 64 + mm) * SP + n] = acc[r] + bias[mm];
  }
}

// ---------------------------------------------------------------------------
// CBAM pooling over the virtual concat [cc | att | x] (256 channels)
// ---------------------------------------------------------------------------
__global__ void k_pool(const float* __restrict__ cc, const float* __restrict__ att,
                       const float* __restrict__ x, float* __restrict__ mx,
                       float* __restrict__ av) {
  int bc = blockIdx.x;                                 // B*256
  int b = bc >> 8, c = bc & 255;
  const float* p = (c < 64)  ? cc  + ((size_t)b * 64 + c) * SP
                 : (c < 128) ? att + ((size_t)b * 64 + (c - 64)) * SP
                             : x   + ((size_t)b * CIN + (c - 128)) * SP;
  int t = threadIdx.x;                                 // 64 threads
  float s = 0.f, m = -3.4e38f;
  for (int i = t; i < SP; i += 64) { float v = p[i]; s += v; m = fmaxf(m, v); }
  __shared__ float ls[64], lm[64];
  ls[t] = s; lm[t] = m;
  __syncthreads();
  for (int off = 32; off > 0; off >>= 1) {
    if (t < off) { ls[t] += ls[t + off]; lm[t] = fmaxf(lm[t], lm[t + off]); }
    __syncthreads();
  }
  if (t == 0) { av[bc] = ls[0] * (1.0f / SP); mx[bc] = lm[0]; }
}

// ---------------------------------------------------------------------------
// CBAM gate: sigmoid( (relu(mx@w1T) + relu(av@w1T)) @ w2T )
// ---------------------------------------------------------------------------
__global__ void k_cbam(const float* __restrict__ mx, const float* __restrict__ av,
                       const float* __restrict__ w1, const float* __restrict__ w2,
                       float* __restrict__ gate) {
  int b = blockIdx.x;
  int t = threadIdx.x;                                 // 256 threads
  __shared__ float hh[32];
  __shared__ float s[16];
  if (t < 32) {
    int h = t & 15;
    const float* v = (t < 16) ? (mx + b * 256) : (av + b * 256);
    float acc = 0.f;
    for (int c = 0; c < 256; ++c) acc += v[c] * w1[h * 256 + c];
    hh[t] = fmaxf(acc, 0.f);
  }
  __syncthreads();
  if (t < 16) s[t] = hh[t] + hh[t + 16];
  __syncthreads();
  float g = 0.f;
  #pragma unroll
  for (int h = 0; h < 16; ++h) g += s[h] * w2[t * 16 + h];
  gate[b * 256 + t] = 1.f / (1.f + expf(-g));
}

// ---------------------------------------------------------------------------
// Final metric conv: M=192, K=256, N=576 with gate fused into the B operand.
// ---------------------------------------------------------------------------
__device__ __forceinline__ float cat_fetch(const float* __restrict__ cc,
                                           const float* __restrict__ att,
                                           const float* __restrict__ x,
                                           int b, int k, int n) {
  if (k < 64)  return cc [((size_t)b * 64  + k) * SP + n];
  if (k < 128) return att[((size_t)b * 64  + (k - 64)) * SP + n];
  return              x  [((size_t)b * CIN + (k - 128)) * SP + n];
}

__global__ void k_metric_wmma(const float* __restrict__ mw, const float* __restrict__ mb,
                              const float* __restrict__ cc, const float* __restrict__ att,
                              const float* __restrict__ x, const float* __restrict__ gate,
                              float* __restrict__ out) {
  const int NT = 36, MT = 12;
  int wave = blockIdx.x * (blockDim.x >> 5) + (threadIdx.x >> 5);
  int lane = threadIdx.x & 31;
  if (wave >= BATCH * MT * NT) return;
  int nt = wave % NT, mt = (wave / NT) % MT, b = wave / (NT * MT);
  int hf = lane >> 4, l = lane & 15;
  int n = nt * 16 + l;
  int m = mt * 16 + l;
  int koff = hf * 2;
  v8f acc = {};
  #pragma unroll 4
  for (int k0 = 0; k0 < 256; k0 += 4) {
    int k = k0 + koff;
    float a0 = mw[(size_t)m * 256 + k];
    float a1 = mw[(size_t)m * 256 + k + 1];
    float v0 = cat_fetch(cc, att, x, b, k, n)     * gate[b * 256 + k];
    float v1 = cat_fetch(cc, att, x, b, k + 1, n) * gate[b * 256 + k + 1];
    acc = wmma4(a0, a1, v0, v1, acc);
  }
  #pragma unroll
  for (int r = 0; r < 8; ++r) {
    int mm = mt * 16 + r + hf * 8;
    out[((size_t)b * 192 + mm) * SP + n] = acc[r] + mb[mm];
  }
}

// ---------------------------------------------------------------------------
extern "C" void kernel_launch(void* const* d_in, const int* in_sizes, int n_in,
                              void* d_out, int out_size, void* d_ws, size_t ws_size,
                              hipStream_t stream) {
  (void)in_sizes; (void)n_in; (void)out_size; (void)ws_size;
  const float* x        = (const float*)d_in[0];
  const float* ref      = (const float*)d_in[1];
  const float* conv1_w  = (const float*)d_in[2];
  const float* conv1_b  = (const float*)d_in[3];
  const float* metric_w = (const float*)d_in[4];
  const float* metric_b = (const float*)d_in[5];
  const float* b1_c1_w  = (const float*)d_in[6];
  const float* b1_c1_b  = (const float*)d_in[7];
  const float* b1_c2_w  = (const float*)d_in[8];
  const float* b1_gn_g  = (const float*)d_in[9];
  const float* b1_gn_b  = (const float*)d_in[10];
  const float* b1_c2_b  = (const float*)d_in[11];
  const float* b2_c1_w  = (const float*)d_in[12];
  const float* b2_c1_b  = (const float*)d_in[13];
  const float* b2_c2_w  = (const float*)d_in[14];
  const float* b2_gn_g  = (const float*)d_in[15];
  const float* b2_gn_b  = (const float*)d_in[16];
  const float* b2_c2_b  = (const float*)d_in[17];
  const float* b3_c1_w  = (const float*)d_in[18];
  const float* b3_c1_b  = (const float*)d_in[19];
  const float* b3_c2_w  = (const float*)d_in[20];
  const float* b3_gn_g  = (const float*)d_in[21];
  const float* b3_gn_b  = (const float*)d_in[22];
  const float* b3_c2_b  = (const float*)d_in[23];
  const float* cbam_w1  = (const float*)d_in[24];
  const float* cbam_w2  = (const float*)d_in[25];

  float* ws = (float*)d_ws;
  size_t off = 0;
  auto alloc = [&](size_t n) { float* p = ws + off; off += n; return p; };
  float* avg   = alloc((size_t)BATCH * CIN);
  float* qn    = alloc((size_t)BATCH * SP * CIN);
  float* sn    = alloc((size_t)BATCH * SP * CIN);
  float* corr  = alloc((size_t)BATCH * SP * SP);
  float* t1    = alloc((size_t)BATCH * SP * 144 * 16);   // channel-last
  float* t2    = alloc((size_t)BATCH * SP * 36 * 64);    // channel-last
  float* t3    = alloc((size_t)BATCH * SP * 36 * 128);   // channel-last
  float* corrm = alloc((size_t)BATCH * CIN * SP);
  float* contr = alloc((size_t)BATCH * 64 * SP);
  float* att   = alloc((size_t)BATCH * 64 * SP);
  float* corrf = alloc((size_t)BATCH * 64 * SP);
  float* ccb   = alloc((size_t)BATCH * 64 * SP);
  float* mx    = alloc((size_t)BATCH * 256);
  float* av    = alloc((size_t)BATCH * 256);
  float* gate  = alloc((size_t)BATCH * 256);
  float* stats = alloc((size_t)BATCH * NGRP * 2);
  float* wt2   = alloc((size_t)2 * 9 * 64 * 16);         // repacked block-2 weights
  float* wt3   = alloc((size_t)2 * 9 * 128 * 64);        // repacked block-3 weights

  // 0. one-shot weight repacks (independent of data path)
  k_repack_w<16, 64><<<(2 * 9 * 64 * 16 + 255) / 256, 256, 0, stream>>>(b2_c1_w, b2_c2_w, wt2);
  k_repack_w<64, 128><<<(2 * 9 * 128 * 64 + 255) / 256, 256, 0, stream>>>(b3_c1_w, b3_c2_w, wt3);
  // 1. reference-frame channel means
  k_avg_ref<<<(BATCH * CIN + 255) / 256, 256, 0, stream>>>(ref, avg);
  // 2. channel-L2 normalize (transposed for GEMM)
  k_normalize<<<BATCH * SP, 128, 0, stream>>>(x, ref, qn, sn);
  // 3. cosine cost volume (WMMA, fused ReLU)
  k_corr_wmma<<<(BATCH * 36 * 36) / 4, 128, 0, stream>>>(qn, sn, corr);
  // 4. block 1 conv4d (ic=1, VALU), channel-last out
  k_conv4d_b1<<<(BATCH * SP * 144 + 255) / 256, 256, 0, stream>>>(
      corr, b1_c1_w, b1_c1_b, b1_c2_w, b1_c2_b, t1);
  {
    size_t per_b = (size_t)SP * 144 * 16, tot = per_b * BATCH;
    k_gn_stats_cl<<<BATCH * NGRP, 256, 0, stream>>>(t1, 16, SP * 144, stats);
    k_gn_apply_cl<<<(unsigned)((tot + 255) / 256), 256, 0, stream>>>(
        t1, 16, b1_gn_g, b1_gn_b, stats, per_b, tot);
  }
  // 5. block 2 conv4d (WMMA): IC=16, OC=64, hb 12 -> 6 (stride 2)
  k_conv4d_wmma<16, 64, 12, 2><<<(BATCH * 4 * ((SP * 36) / 16)) / 4, 128, 0, stream>>>(
      t1, t2, wt2, b2_c1_b, b2_c2_b);
  {
    size_t per_b = (size_t)SP * 36 * 64, tot = per_b * BATCH;
    k_gn_stats_cl<<<BATCH * NGRP, 256, 0, stream>>>(t2, 64, SP * 36, stats);
    k_gn_apply_cl<<<(unsigned)((tot + 255) / 256), 256, 0, stream>>>(
        t2, 64, b2_gn_g, b2_gn_b, stats, per_b, tot);
  }
  // 6. block 3 conv4d (WMMA): IC=64, OC=128, hb 6 -> 6 (stride 1)
  k_conv4d_wmma<64, 128, 6, 1><<<(BATCH * 8 * ((SP * 36) / 16)) / 4, 128, 0, stream>>>(
      t2, t3, wt3, b3_c1_b, b3_c2_b);
  {
    size_t per_b = (size_t)SP * 36 * 128, tot = per_b * BATCH;
    k_gn_stats_cl<<<BATCH * NGRP, 256, 0, stream>>>(t3, 128, SP * 36, stats);
    k_gn_apply_cl<<<(unsigned)((tot + 255) / 256), 256, 0, stream>>>(
        t3, 128, b3_gn_g, b3_gn_b, stats, per_b, tot);
  }
  // 7. mean over (hb', wb') + transpose back to channel-first
  k_mean_bb_cl<<<BATCH * SP, 128, 0, stream>>>(t3, corrm);
  // 8. pointwise conv GEMMs (WMMA, fused elementwise prep)
  k_pw<1><<<576, 128, 0, stream>>>(conv1_w, conv1_b, avg,   x,     contr); // |avg_ref - x|
  k_pw<2><<<576, 128, 0, stream>>>(conv1_w, conv1_b, corrm, x,     att);   // |corr - x|
  k_pw<0><<<576, 128, 0, stream>>>(conv1_w, conv1_b, corrm, corrm, corrf); // corr
  k_pw<3><<<576, 128, 0, stream>>>(conv1_w, conv1_b, corrf, contr, ccb);   // concat
  // 9. CBAM pooling + gate
  k_pool<<<BATCH * 256, 64, 0, stream>>>(ccb, att, x, mx, av);
  k_cbam<<<BATCH, 256, 0, stream>>>(mx, av, cbam_w1, cbam_w2, gate);
  // 10. gated metric conv (WMMA) -> d_out [B,192,24,24]
  k_metric_wmma<<<(BATCH * 12 * 36) / 4, 128, 0, stream>>>(
      metric_w, metric_b, ccb, att, x, gate, (float*)d_out);
}